// Stitching_circle_79826262163495
// MI455X (gfx1250) — compile-verified
//
#include <hip/hip_runtime.h>

// Problem constants (match reference: B, C, H, W = 64, 3, 512, 512)
#define BATCH 64
#define CH    3
#define HDIM  512
#define WDIM  512
#define PLANE (HDIM * WDIM)          // 262144 elements per channel plane

typedef float float4v __attribute__((ext_vector_type(4)));

// One block = one (batch, row). 128 threads * 4 cols = 512 = WDIM.
// wave32: 128 threads = 4 waves.
__global__ __launch_bounds__(128) void Stitching_circle_blend_kernel(
    const float* __restrict__ x,
    const float* __restrict__ ori,
    const float* __restrict__ bbox,
    float* __restrict__ out)
{
    const int row = blockIdx.x & (HDIM - 1);
    const int b   = blockIdx.x >> 9;          // HDIM == 512

    // --- per-block (uniform) ellipse parameters: scalar path ---
    const float bx1 = bbox[b * 5 + 0];
    const float by1 = bbox[b * 5 + 1];
    const float bx2 = bbox[b * 5 + 2];
    const float by2 = bbox[b * 5 + 3];
    // faithful to reference: cols 0,2 scaled by h; cols 1,3 by w; trunc toward zero
    const int x1 = (int)(bx1 * (float)HDIM);
    const int y1 = (int)(by1 * (float)WDIM);
    const int x2 = (int)(bx2 * (float)HDIM);
    const int y2 = (int)(by2 * (float)WDIM);
    const int cx = (x1 + x2) >> 1;            // positive -> floor-div == >>1
    const int cy = (y1 + y2) >> 1;
    // 9/16 and 5/8 are exact powers-of-two multiples; rintf == round-half-even
    const float l = rintf((float)(y2 - y1) * (9.0f / 16.0f)); // semi-axis along rows
    const float s = rintf((float)(x2 - x1) * (5.0f / 8.0f));  // semi-axis along cols

    // IEEE divide to match jnp exactly at the ellipse boundary
    const float dr  = ((float)row - (float)cy) / l;
    const float dr2 = dr * dr;

    // --- per-thread mask for 4 consecutive columns ---
    const int col0 = (int)threadIdx.x * 4;
    bool ins[4];
    bool all_in = true, any_in = false;
#pragma unroll
    for (int i = 0; i < 4; ++i) {
        const float dc = ((float)(col0 + i) - (float)cx) / s;
        ins[i] = (dr2 + dc * dc) <= 1.0f;
        all_in = all_in && ins[i];
        any_in = any_in || ins[i];
    }

    const long long rowbase = ((long long)b * CH * HDIM + row) * WDIM + col0;

    if (all_in) {
        // Fully inside: out == x exactly; skip reading ori (~15% of pixels)
#pragma unroll
        for (int c = 0; c < CH; ++c) {
            const long long off = rowbase + (long long)c * PLANE;
            float4v xv = __builtin_nontemporal_load((const float4v*)(x + off));
            __builtin_nontemporal_store(xv, (float4v*)(out + off));
        }
    } else if (!any_in) {
        // Fully outside: out == ori exactly; skip reading x (~85% of pixels)
#pragma unroll
        for (int c = 0; c < CH; ++c) {
            const long long off = rowbase + (long long)c * PLANE;
            float4v ov = __builtin_nontemporal_load((const float4v*)(ori + off));
            __builtin_nontemporal_store(ov, (float4v*)(out + off));
        }
    } else {
        // Boundary cachelines only: read both, per-lane select (exact blend)
#pragma unroll
        for (int c = 0; c < CH; ++c) {
            const long long off = rowbase + (long long)c * PLANE;
            float4v xv = __builtin_nontemporal_load((const float4v*)(x + off));
            float4v ov = __builtin_nontemporal_load((const float4v*)(ori + off));
            float4v r;
#pragma unroll
            for (int i = 0; i < 4; ++i) r[i] = ins[i] ? xv[i] : ov[i];
            __builtin_nontemporal_store(r, (float4v*)(out + off));
        }
    }
}

extern "C" void kernel_launch(void* const* d_in, const int* in_sizes, int n_in,
                              void* d_out, int out_size, void* d_ws, size_t ws_size,
                              hipStream_t stream) {
    (void)in_sizes; (void)n_in; (void)out_size; (void)d_ws; (void)ws_size;
    const float* x    = (const float*)d_in[0];
    const float* ori  = (const float*)d_in[1];
    const float* bbox = (const float*)d_in[2];
    float* out        = (float*)d_out;

    // one block per (batch, row): 64*512 = 32768 blocks of 128 threads (4 wave32)
    dim3 grid(BATCH * HDIM);
    dim3 block(128);
    Stitching_circle_blend_kernel<<<grid, block, 0, stream>>>(x, ori, bbox, out);
}